// HMM_71210557768002
// MI455X (gfx1250) — compile-verified
//
#include <hip/hip_runtime.h>
#include <math.h>

// HMM forward-backward (log-space) for B=32, T=8192, K=64.
// Strategy: latency-bound serial recurrence -> per-step 16x64x64 SGEMM via
// V_WMMA_F32_16X16X4_F32, forward and backward run concurrently (4 blocks).

typedef __attribute__((ext_vector_type(2))) float v2f;
typedef __attribute__((ext_vector_type(8))) float v8f;

#define HMM_B 32
#define HMM_T 8192
#define HMM_K 64
#define NEG_HALF_LOG_2PI (-0.91893853320467274178f)

__device__ __forceinline__ v8f wmma_f32(v2f a, v2f b, v8f c) {
  // D = A(16x4 f32) * B(4x16 f32) + C(16x16 f32)
  return __builtin_amdgcn_wmma_f32_16x16x4_f32(
      /*neg_a=*/false, a, /*neg_b=*/false, b,
      /*c_mod=*/(short)0, c, /*reuse_a=*/false, /*reuse_b=*/false);
}

// Blocks 0,1: forward recurrence over batch rows [0,16) / [16,32) -> fs (d_ws)
// Blocks 2,3: backward recurrence (same split)                   -> raw b (d_out)
__global__ __launch_bounds__(32)
void hmm_recur_kernel(const float* __restrict__ obvs,        // [B][T]
                      const float* __restrict__ log_init,    // [K]
                      const float* __restrict__ transfer,    // [K][K]
                      const float* __restrict__ means,       // [K]
                      const float* __restrict__ log_stds,    // [K]
                      float* __restrict__ fs,                // [T][B][K] (ws)
                      float* __restrict__ out)               // [B][T][K] raw b
{
  __shared__ float lp[16][66];   // p = exp(v - rowmax), padded stride (even)
  __shared__ float lx[16];       // obvs staging for current t

  const int  lane   = threadIdx.x;      // 0..31 (wave32)
  const int  half   = lane >> 4;        // 0 | 1
  const int  l16    = lane & 15;
  const bool is_bwd = (blockIdx.x >= 2);
  const int  base   = (blockIdx.x & 1) * 16;   // batch-row base

  // Per-lane constants for its 4 k-columns (k = nt*16 + l16)
  float mean_[4], istd_[4], nls_[4], lini_[4];
#pragma unroll
  for (int nt = 0; nt < 4; ++nt) {
    int k = nt * 16 + l16;
    float ls  = log_stds[k];
    mean_[nt] = means[k];
    istd_[nt] = __expf(-ls);
    nls_[nt]  = -ls + NEG_HALF_LOG_2PI;
    lini_[nt] = log_init[k];
  }

  // B fragments (constant all steps): B[j][k] = transfer[k][j].
  // f32 16x4-style B layout: lanes 0-15 rows {4kk+0,4kk+1}, lanes 16-31 rows {4kk+2,4kk+3}.
  v2f bfr[16][4];
#pragma unroll
  for (int kk = 0; kk < 16; ++kk)
#pragma unroll
    for (int nt = 0; nt < 4; ++nt) {
      int j0 = kk * 4 + half * 2;
      int kc = nt * 16 + l16;
      v2f b;
      b.x = transfer[kc * HMM_K + j0];
      b.y = transfer[kc * HMM_K + j0 + 1];
      bfr[kk][nt] = b;
    }

  float rmax[8];       // row max for row (half ? v+8 : v)
  float cur[4][8];     // current f (fwd) or b_t + e_t (bwd), D layout

  // ---------------- initial step ----------------
  const int t0 = is_bwd ? (HMM_T - 1) : 0;
  if (lane < 16) lx[lane] = obvs[(size_t)(base + lane) * HMM_T + t0];
  __syncthreads();

#pragma unroll
  for (int nt = 0; nt < 4; ++nt)
#pragma unroll
    for (int v = 0; v < 8; ++v) {
      int row = half ? v + 8 : v;
      int bg  = base + row;
      int k   = nt * 16 + l16;
      float x = lx[row];
      float z = (x - mean_[nt]) * istd_[nt];
      float e = __fmaf_rn(-0.5f * z, z, nls_[nt]);
      if (is_bwd) {
        out[((size_t)bg * HMM_T + (HMM_T - 1)) * HMM_K + k] = lini_[nt]; // b_{T-1}
        cur[nt][v] = lini_[nt] + e;                                      // v = b+e
      } else {
        float f = e + lini_[nt];                                         // f_0
        fs[((size_t)t0 * HMM_B + bg) * HMM_K + k] = f;
        cur[nt][v] = f;
      }
    }

  // rowmax + p into LDS
#pragma unroll
  for (int v = 0; v < 8; ++v) {
    float r = fmaxf(fmaxf(cur[0][v], cur[1][v]), fmaxf(cur[2][v], cur[3][v]));
    r = fmaxf(r, __shfl_xor(r, 1));
    r = fmaxf(r, __shfl_xor(r, 2));
    r = fmaxf(r, __shfl_xor(r, 4));
    r = fmaxf(r, __shfl_xor(r, 8));
    rmax[v] = r;
    int row = half ? v + 8 : v;
#pragma unroll
    for (int nt = 0; nt < 4; ++nt)
      lp[row][nt * 16 + l16] = __expf(cur[nt][v] - r);
  }

  // ---------------- main recurrence ----------------
  const int tsteps = HMM_T - 1;
  for (int it = 1; it <= tsteps; ++it) {
    int t = is_bwd ? (HMM_T - 1 - it) : it;
    __syncthreads();  // single-wave WG: cheap; orders LDS p/x traffic

    // s = p @ transfer^T : 4 N-tiles, 16 K-chunks of 4
    v8f acc[4] = {};
#pragma unroll
    for (int kk = 0; kk < 16; ++kk) {
      // A fragment: rows = l16 for ALL lanes; cols kk*4 + half*2 + {0,1}
      v2f a = *(const v2f*)&lp[l16][kk * 4 + half * 2];
      acc[0] = wmma_f32(a, bfr[kk][0], acc[0]);
      acc[1] = wmma_f32(a, bfr[kk][1], acc[1]);
      acc[2] = wmma_f32(a, bfr[kk][2], acc[2]);
      acc[3] = wmma_f32(a, bfr[kk][3], acc[3]);
    }

    // stage obvs for this t (in-order LDS within the wave)
    if (lane < 16) lx[lane] = obvs[(size_t)(base + lane) * HMM_T + t];

#pragma unroll
    for (int nt = 0; nt < 4; ++nt)
#pragma unroll
      for (int v = 0; v < 8; ++v) {
        int row = half ? v + 8 : v;
        int bg  = base + row;
        int k   = nt * 16 + l16;
        float val = __logf(acc[nt][v]) + rmax[v];   // log-matvec + m_prev
        float x = lx[row];
        float z = (x - mean_[nt]) * istd_[nt];
        float e = __fmaf_rn(-0.5f * z, z, nls_[nt]);
        if (is_bwd) {
          out[((size_t)bg * HMM_T + t) * HMM_K + k] = val;  // raw b_t
          cur[nt][v] = val + e;                             // v = b_t + e_t
        } else {
          float f = val + e;                                // f_t = e_t + lme
          fs[((size_t)t * HMM_B + bg) * HMM_K + k] = f;
          cur[nt][v] = f;
        }
      }

    if (it == tsteps) break;   // no p needed after last step (uniform)

#pragma unroll
    for (int v = 0; v < 8; ++v) {
      float r = fmaxf(fmaxf(cur[0][v], cur[1][v]), fmaxf(cur[2][v], cur[3][v]));
      r = fmaxf(r, __shfl_xor(r, 1));
      r = fmaxf(r, __shfl_xor(r, 2));
      r = fmaxf(r, __shfl_xor(r, 4));
      r = fmaxf(r, __shfl_xor(r, 8));
      rmax[v] = r;
      int row = half ? v + 8 : v;
#pragma unroll
      for (int nt = 0; nt < 4; ++nt)
        lp[row][nt * 16 + l16] = __expf(cur[nt][v] - r);
    }
  }
}

// gamma = fs + b_raw, row-LSE normalized, in place over d_out.
// One wave per (b,t) row; lane owns k = lane and lane+32.
__global__ __launch_bounds__(256)
void hmm_gamma_kernel(const float* __restrict__ fs, float* __restrict__ out) {
  int lane = threadIdx.x & 31;
  int wave = threadIdx.x >> 5;
  int row  = blockIdx.x * 8 + wave;        // [0, B*T)
  int b = row >> 13;                       // T = 8192
  int t = row & (HMM_T - 1);
  size_t fbase = ((size_t)t * HMM_B + b) * HMM_K;
  size_t obase = ((size_t)b * HMM_T + t) * HMM_K;
  float g0 = fs[fbase + lane]      + out[obase + lane];
  float g1 = fs[fbase + lane + 32] + out[obase + lane + 32];
  float mx = fmaxf(g0, g1);
#pragma unroll
  for (int m = 1; m < 32; m <<= 1) mx = fmaxf(mx, __shfl_xor(mx, m));
  float s = __expf(g0 - mx) + __expf(g1 - mx);
#pragma unroll
  for (int m = 1; m < 32; m <<= 1) s += __shfl_xor(s, m);
  float lse = mx + __logf(s);
  out[obase + lane]      = g0 - lse;
  out[obase + lane + 32] = g1 - lse;
}

extern "C" void kernel_launch(void* const* d_in, const int* in_sizes, int n_in,
                              void* d_out, int out_size, void* d_ws, size_t ws_size,
                              hipStream_t stream) {
  const float* obvs     = (const float*)d_in[0];   // [B*T]
  const float* log_init = (const float*)d_in[1];   // [K]
  const float* transfer = (const float*)d_in[2];   // [K*K]
  const float* means    = (const float*)d_in[3];   // [K]
  const float* log_stds = (const float*)d_in[4];   // [K]
  float* out = (float*)d_out;                      // [B*T*K]
  float* fs  = (float*)d_ws;                       // [T*B*K] = 64 MB

  hmm_recur_kernel<<<dim3(4), dim3(32), 0, stream>>>(
      obvs, log_init, transfer, means, log_stds, fs, out);
  hmm_gamma_kernel<<<dim3((HMM_B * HMM_T) / 8), dim3(256), 0, stream>>>(fs, out);

  (void)in_sizes; (void)n_in; (void)out_size; (void)ws_size;
}